// LightGCN_64192581206146
// MI455X (gfx1250) — compile-verified
//
#include <hip/hip_runtime.h>
#include <hip/hip_bf16.h>

// ---------------- problem constants (match reference) ----------------
#define NUM_USERS 100000
#define NUM_ITEMS 50000
#define NN        (NUM_USERS + NUM_ITEMS)   // 150000 nodes
#define E_DIR     600000
#define EE        (2 * E_DIR)               // 1,200,000 edges (multiple of 16)
#define DD        64
#define F_EDGE    8
#define H_MLP     32
#define NUM_LAYERS 3
#define AUDIO_SCALE 0.3f
#define META_SCALE  0.44f

typedef __attribute__((ext_vector_type(2))) float v2f;
typedef __attribute__((ext_vector_type(8))) float v8f;

// ---------------- helpers ----------------
__device__ __forceinline__ float wave_reduce_sum32(float v) {
    // full 32-lane (wave32) butterfly reduction
    #pragma unroll
    for (int m = 16; m >= 1; m >>= 1) v += __shfl_xor(v, m, 32);
    return v;
}

// ---------------- zero a float buffer (grid-stride) ----------------
__global__ void k_zero(float* __restrict__ p, size_t n) {
    size_t i = (size_t)blockIdx.x * blockDim.x + threadIdx.x;
    size_t stride = (size_t)gridDim.x * blockDim.x;
    for (; i < n; i += stride) p[i] = 0.0f;
}

// ---------------- node init: x0 rows, also seeds acc ----------------
// one wave32 per node; lane handles 2 consecutive floats of the 64-wide row
__global__ __launch_bounds__(256) void k_node_init(
    const float* __restrict__ user_w, const float* __restrict__ audio,
    const float* __restrict__ artist_w, const float* __restrict__ album_w,
    const int* __restrict__ artist_ids, const int* __restrict__ album_ids,
    float* __restrict__ x, float* __restrict__ acc)
{
    int wave = blockIdx.x * (blockDim.x >> 5) + (threadIdx.x >> 5);
    int lane = threadIdx.x & 31;
    if (wave >= NN) return;
    float2 v;
    if (wave < NUM_USERS) {
        v = *(const float2*)(user_w + (size_t)wave * DD + 2 * lane);
    } else {
        int i = wave - NUM_USERS;
        float2 a  = *(const float2*)(audio    + (size_t)i * DD + 2 * lane);
        int ar = artist_ids[i];
        int al = album_ids[i];
        float2 m1 = *(const float2*)(artist_w + (size_t)ar * DD + 2 * lane);
        float2 m2 = *(const float2*)(album_w  + (size_t)al * DD + 2 * lane);
        v.x = a.x * AUDIO_SCALE + (m1.x + m2.x) * META_SCALE;
        v.y = a.y * AUDIO_SCALE + (m1.y + m2.y) * META_SCALE;
    }
    float ss = wave_reduce_sum32(v.x * v.x + v.y * v.y);
    float inv = 1.0f / fmaxf(sqrtf(ss), 1e-12f);
    v.x *= inv; v.y *= inv;
    *(float2*)(x   + (size_t)wave * DD + 2 * lane) = v;
    *(float2*)(acc + (size_t)wave * DD + 2 * lane) = v;
}

// ---------------- edge MLP via V_WMMA_F32_16X16X4_F32 ----------------
// One wave32 per 16-edge tile.  (16x8 edge-features) @ (8x32 W1) is computed
// as 2 N-tiles x 2 K-slices of 16x16x4 f32 WMMA, then bias+ReLU, then the
// 32->1 second layer is folded into per-lane partials reduced across the
// 16-lane half-wave.  EE % 16 == 0 so EXEC is always all-ones here.
//
// f32 A-matrix 16x4 layout: lanes 0-15 hold rows M=0..15 with K={0,1} in the
// two VGPRs; lanes 16-31 hold K={2,3}.  B (4x16) mirrors with N striped on
// lanes.  C/D 16x16: lane holds column N=lane%16; VGPR r = row r (+8 for the
// upper half-wave).
__global__ __launch_bounds__(256) void k_edge_mlp(
    const float* __restrict__ ef,  // (E, 8)
    const float* __restrict__ w1,  // (8, 32)
    const float* __restrict__ b1,  // (32)
    const float* __restrict__ w2,  // (32, 1)
    const float* __restrict__ b2,  // (1)
    float* __restrict__ edge_w)    // (E)
{
    int wave = blockIdx.x * (blockDim.x >> 5) + (threadIdx.x >> 5);
    int lane = threadIdx.x & 31;
    int half = lane >> 4;          // 0: K={0,1}(+4) | 1: K={2,3}(+4)
    int sub  = lane & 15;
    int e0   = wave * 16;
    if (e0 >= EE) return;          // wave-uniform; never taken with exact grid

    const float* efr = ef + (size_t)(e0 + sub) * F_EDGE + half * 2;
    v2f a0; a0.x = efr[0]; a0.y = efr[1];   // K-slice 0
    v2f a1; a1.x = efr[4]; a1.y = efr[5];   // K-slice 1

    float bias2 = b2[0];
    float part[8];
    #pragma unroll
    for (int r = 0; r < 8; ++r) part[r] = 0.0f;

    #pragma unroll
    for (int nt = 0; nt < 2; ++nt) {
        int n = nt * 16 + sub;  // this lane's output column of H_MLP
        v2f bb0, bb1;
        bb0.x = w1[(half * 2 + 0) * H_MLP + n];
        bb0.y = w1[(half * 2 + 1) * H_MLP + n];
        bb1.x = w1[(half * 2 + 4) * H_MLP + n];
        bb1.y = w1[(half * 2 + 5) * H_MLP + n];
        v8f c = {0.f, 0.f, 0.f, 0.f, 0.f, 0.f, 0.f, 0.f};
        c = __builtin_amdgcn_wmma_f32_16x16x4_f32(false, a0, false, bb0,
                                                  (short)0, c, false, false);
        c = __builtin_amdgcn_wmma_f32_16x16x4_f32(false, a1, false, bb1,
                                                  (short)0, c, false, false);
        float bn = b1[n];
        float wn = w2[n];
        #pragma unroll
        for (int r = 0; r < 8; ++r) {
            float h = c[r] + bn;
            h = h > 0.0f ? h : 0.0f;      // ReLU
            part[r] += h * wn;            // fold in second-layer matvec
        }
    }
    // reduce the 32-wide hidden dim across the 16 lanes of each half-wave
    #pragma unroll
    for (int m = 1; m < 16; m <<= 1) {
        #pragma unroll
        for (int r = 0; r < 8; ++r) part[r] += __shfl_xor(part[r], m, 32);
    }
    if (sub == 0) {
        #pragma unroll
        for (int r = 0; r < 8; ++r) {
            float z = part[r] + bias2;
            edge_w[e0 + half * 8 + r] = 1.0f / (1.0f + __expf(-z)); // sigmoid
        }
    }
}

// ---------------- degree accumulation ----------------
__global__ void k_degree(const int* __restrict__ col,
                         const float* __restrict__ ew,
                         float* __restrict__ deg)
{
    int e = blockIdx.x * blockDim.x + threadIdx.x;
    if (e < EE) atomicAdd(&deg[col[e]], ew[e]);
}

// ---------------- dinv = deg>0 ? 1/sqrt(deg) : 0 ----------------
__global__ void k_dinv(const float* __restrict__ deg, float* __restrict__ dinv)
{
    int n = blockIdx.x * blockDim.x + threadIdx.x;
    if (n < NN) {
        float d = deg[n];
        dinv[n] = d > 0.0f ? (1.0f / sqrtf(d)) : 0.0f;
    }
}

// ---------------- per-edge propagation coefficient ----------------
__global__ void k_coef(const int* __restrict__ row, const int* __restrict__ col,
                       const float* __restrict__ ew, const float* __restrict__ dinv,
                       float* __restrict__ nrm)
{
    int e = blockIdx.x * blockDim.x + threadIdx.x;
    if (e < EE) nrm[e] = dinv[row[e]] * ew[e] * dinv[col[e]];
}

// ---------------- one propagation layer: x_out[col] += nrm * x_in[row] ----
// Block stages 256 edge records into LDS (coalesced); each wave then streams
// 32 edges, lane covers 2 floats of the 64-wide row: float2 gather from L2
// (x fits in the 192MB L2) + two global f32 atomic adds.
__global__ __launch_bounds__(256) void k_spmm(
    const int* __restrict__ row, const int* __restrict__ col,
    const float* __restrict__ nrm,
    const float* __restrict__ xin, float* __restrict__ xout)
{
    __shared__ int   srow[256];
    __shared__ int   scol[256];
    __shared__ float snrm[256];
    int tid = threadIdx.x;
    int e0  = blockIdx.x * 256;
    int e   = e0 + tid;
    if (e < EE) { srow[tid] = row[e]; scol[tid] = col[e]; snrm[tid] = nrm[e]; }
    __syncthreads();

    int wave = tid >> 5;
    int lane = tid & 31;
    int base = wave * 32;
    #pragma unroll 4
    for (int k = 0; k < 32; ++k) {
        int ee = e0 + base + k;
        if (ee >= EE) break;
        int   r = srow[base + k];
        int   c = scol[base + k];
        float w = snrm[base + k];
        float2 v = *(const float2*)(xin + (size_t)r * DD + 2 * lane);
        float* dst = xout + (size_t)c * DD + 2 * lane;
        atomicAdd(dst + 0, w * v.x);
        atomicAdd(dst + 1, w * v.y);
    }
}

// ---------------- acc += x ----------------
__global__ void k_add(float* __restrict__ acc, const float* __restrict__ x, size_t n)
{
    size_t i = (size_t)blockIdx.x * blockDim.x + threadIdx.x;
    size_t stride = (size_t)gridDim.x * blockDim.x;
    for (; i < n; i += stride) acc[i] += x[i];
}

// ---------------- finalize: out = l2norm(acc/4); align_loss = 0 ----------
__global__ __launch_bounds__(256) void k_final(const float* __restrict__ acc,
                                               float* __restrict__ out)
{
    int wave = blockIdx.x * (blockDim.x >> 5) + (threadIdx.x >> 5);
    int lane = threadIdx.x & 31;
    if (blockIdx.x == 0 && threadIdx.x == 0) out[(size_t)NN * DD] = 0.0f;
    if (wave >= NN) return;
    float2 v = *(const float2*)(acc + (size_t)wave * DD + 2 * lane);
    v.x *= 0.25f; v.y *= 0.25f;   // / (NUM_LAYERS + 1)
    float ss = wave_reduce_sum32(v.x * v.x + v.y * v.y);
    float inv = 1.0f / fmaxf(sqrtf(ss), 1e-12f);
    v.x *= inv; v.y *= inv;
    *(float2*)(out + (size_t)wave * DD + 2 * lane) = v;
}

// ---------------- host orchestration ----------------
extern "C" void kernel_launch(void* const* d_in, const int* in_sizes, int n_in,
                              void* d_out, int out_size, void* d_ws, size_t ws_size,
                              hipStream_t stream)
{
    (void)in_sizes; (void)n_in; (void)out_size; (void)ws_size;
    const float* user_w  = (const float*)d_in[0];
    const float* audio   = (const float*)d_in[1];
    const float* artistw = (const float*)d_in[2];
    const float* albumw  = (const float*)d_in[3];
    const float* w1      = (const float*)d_in[4];
    const float* b1      = (const float*)d_in[5];
    const float* w2      = (const float*)d_in[6];
    const float* b2      = (const float*)d_in[7];
    const float* ef      = (const float*)d_in[8];
    const int*   eidx    = (const int*)d_in[9];
    const int*   arids   = (const int*)d_in[10];
    const int*   alids   = (const int*)d_in[11];
    const int* row = eidx;          // edge_index[0]
    const int* col = eidx + EE;     // edge_index[1]
    float* out = (float*)d_out;

    // workspace layout (floats)
    float* ws   = (float*)d_ws;
    const size_t ND = (size_t)NN * DD;         // 9.6M
    float* xA   = ws;                          // current x
    float* xB   = xA  + ND;                    // next x
    float* acc  = xB  + ND;                    // accumulator
    float* ew   = acc + ND;                    // edge weights (E)
    float* nrm  = ew  + EE;                    // propagation coeff (E)
    float* deg  = nrm + EE;                    // degrees (N)
    float* dinv = deg + NN;                    // 1/sqrt(deg) (N)

    const int TB = 256;
    // node init (writes xA and acc)
    {
        dim3 g((NN + 7) / 8);
        k_node_init<<<g, TB, 0, stream>>>(user_w, audio, artistw, albumw,
                                          arids, alids, xA, acc);
    }
    // edge MLP (WMMA) -> ew
    {
        dim3 g((EE / 16 + 7) / 8);   // 9375 blocks, 8 tiles each, exact
        k_edge_mlp<<<g, TB, 0, stream>>>(ef, w1, b1, w2, b2, ew);
    }
    // degrees
    k_zero<<<dim3(1024), TB, 0, stream>>>(deg, (size_t)NN);
    k_degree<<<dim3((EE + TB - 1) / TB), TB, 0, stream>>>(col, ew, deg);
    k_dinv<<<dim3((NN + TB - 1) / TB), TB, 0, stream>>>(deg, dinv);
    k_coef<<<dim3((EE + TB - 1) / TB), TB, 0, stream>>>(row, col, ew, dinv, nrm);

    // 3 propagation layers
    float* xin = xA;
    float* xout = xB;
    for (int l = 0; l < NUM_LAYERS; ++l) {
        k_zero<<<dim3(4096), TB, 0, stream>>>(xout, ND);
        k_spmm<<<dim3((EE + 255) / 256), TB, 0, stream>>>(row, col, nrm, xin, xout);
        k_add<<<dim3(4096), TB, 0, stream>>>(acc, xout, ND);
        float* t = xin; xin = xout; xout = t;
    }

    // finalize into d_out (user rows then item rows, then align_loss scalar)
    {
        dim3 g((NN + 7) / 8);
        k_final<<<g, TB, 0, stream>>>(acc, out);
    }
}